// MoE_61100204753332
// MI455X (gfx1250) — compile-verified
//
#include <hip/hip_runtime.h>
#include <math.h>

#define BATCH 8192
#define DIN   1024
#define DOUT  1024
#define NEXP  16
#define CAP   BATCH            // worst case: every token picks this expert

typedef __attribute__((ext_vector_type(2))) float v2f;
typedef __attribute__((ext_vector_type(8))) float v8f;
typedef __attribute__((ext_vector_type(4))) int   v4i;

// ---- CDNA5 async global->LDS path (ASYNCcnt), with portable fallback ----
#if defined(__has_builtin)
#if __has_builtin(__builtin_amdgcn_global_load_async_to_lds_b128) && \
    __has_builtin(__builtin_amdgcn_s_wait_asynccnt)
#define MOE_ASYNC 1
#endif
#endif

// pointee-qualified pointer types matching the builtin's signature:
// (v4i addrspace(1)* gaddr, v4i addrspace(3)* lds, imm offset, imm cpol)
typedef __attribute__((address_space(1))) v4i* gv4i_p;
typedef __attribute__((address_space(3))) v4i* lv4i_p;

__device__ __forceinline__ void cp16(float* lds, const float* g) {
#ifdef MOE_ASYNC
    __builtin_amdgcn_global_load_async_to_lds_b128(
        (gv4i_p)(uintptr_t)g,
        (lv4i_p)(unsigned int)(uintptr_t)lds, 0, 0);
#else
    *(float4*)lds = *(const float4*)g;
#endif
}
#ifdef MOE_ASYNC
#define ASYNC_WAIT(N) __builtin_amdgcn_s_wait_asynccnt(N)
#else
#define ASYNC_WAIT(N)
#endif

// ---------------------------------------------------------------- zero ws
__global__ void k_zero(int* counts, float* eload) {
    int t = threadIdx.x;
    if (t < NEXP) { counts[t] = 0; eload[t] = 0.0f; }
}

// ---------------------------------------------------------------- gating
__global__ __launch_bounds__(256)
void k_gate(const float* __restrict__ x, const float* __restrict__ Wg,
            const float* __restrict__ bg,
            int* __restrict__ tkidx, float* __restrict__ tkp,
            int* __restrict__ counts, float* __restrict__ eload,
            int* __restrict__ perm, float* __restrict__ pprob) {
    const int t = blockIdx.x * blockDim.x + threadIdx.x;
    float acc[NEXP];
#pragma unroll
    for (int e = 0; e < NEXP; ++e) acc[e] = bg[e];

    const float4* xr = (const float4*)(x + (size_t)t * DIN);
    for (int k4 = 0; k4 < DIN / 4; ++k4) {
        float4 xv = xr[k4];
#pragma unroll
        for (int e = 0; e < NEXP; ++e) {
            float4 wv = *(const float4*)(Wg + (size_t)e * DIN + k4 * 4);
            acc[e] += xv.x * wv.x + xv.y * wv.y + xv.z * wv.z + xv.w * wv.w;
        }
    }

    int   i0 = 0;  float v0 = acc[0];
    int   i1 = -1; float v1 = -INFINITY;
#pragma unroll
    for (int e = 1; e < NEXP; ++e) {
        float v = acc[e];
        if (v > v0)      { v1 = v0; i1 = i0; v0 = v; i0 = e; }
        else if (v > v1) { v1 = v;  i1 = e; }
    }
    float e1 = __expf(v1 - v0);
    float inv = 1.0f / (1.0f + e1);
    float p0 = inv, p1 = e1 * inv;

    tkidx[t * 2 + 0] = i0;  tkidx[t * 2 + 1] = i1;
    tkp[t * 2 + 0]   = p0;  tkp[t * 2 + 1]   = p1;

    atomicAdd(eload + i0, p0);
    atomicAdd(eload + i1, p1);

    int s0 = atomicAdd(counts + i0, 1);
    perm[i0 * CAP + s0] = t;  pprob[i0 * CAP + s0] = p0;
    int s1 = atomicAdd(counts + i1, 1);
    perm[i1 * CAP + s1] = t;  pprob[i1 * CAP + s1] = p1;
}

// ---------------------------------------------------------------- pad buckets to /128
__global__ void k_pad(const int* __restrict__ counts,
                      int* __restrict__ perm, float* __restrict__ pprob) {
    const int e   = blockIdx.x;
    const int cnt = counts[e];
    const int end = (cnt + 127) & ~127;
    for (int s = cnt + threadIdx.x; s < end; s += 128) {
        perm[e * CAP + s]  = 0;
        pprob[e * CAP + s] = 0.0f;   // p=0 => contributes nothing
    }
}

// ---------------------------------------------------------------- output = weighted bias
__global__ __launch_bounds__(256)
void k_init(const float* __restrict__ be, const int* __restrict__ tkidx,
            const float* __restrict__ tkp, float* __restrict__ out) {
    const int t  = blockIdx.x;
    const int i0 = tkidx[t * 2 + 0], i1 = tkidx[t * 2 + 1];
    const float p0 = tkp[t * 2 + 0], p1 = tkp[t * 2 + 1];
    const int c4 = threadIdx.x;                       // 256 float4 per row
    float4 b0 = ((const float4*)(be + (size_t)i0 * DOUT))[c4];
    float4 b1 = ((const float4*)(be + (size_t)i1 * DOUT))[c4];
    float4 o;
    o.x = p0 * b0.x + p1 * b1.x;
    o.y = p0 * b0.y + p1 * b1.y;
    o.z = p0 * b0.z + p1 * b1.z;
    o.w = p0 * b0.w + p1 * b1.w;
    ((float4*)(out + (size_t)t * DOUT))[c4] = o;
}

// ---------------------------------------------------------------- grouped expert GEMM
// Block tile 128 tokens x 128 cols, K chunked by 16, double-buffered LDS filled
// by GLOBAL_LOAD_ASYNC_TO_LDS_B128. 8 waves (wave32): wave tile 32x64 =
// 2x4 16x16 f32 C tiles via V_WMMA_F32_16X16X4_F32.
#define BM   128
#define BN   128
#define KC   16
#define LDA  20                 // 80B row stride: 16B aligned, conflict-free b64 reads
#define BUFE (BM * LDA)         // floats per buffer

__global__ __launch_bounds__(256)
void k_gemm(const float* __restrict__ x, const float* __restrict__ We,
            const int* __restrict__ counts, const int* __restrict__ perm,
            const float* __restrict__ pprob, float* __restrict__ out) {
    __shared__ __align__(16) float As[2 * BUFE];
    __shared__ __align__(16) float Bs[2 * BUFE];
    __shared__ int   tok[BM];
    __shared__ float pr[BM];

    const int e   = blockIdx.z;
    const int cnt = counts[e];
    const int m0  = blockIdx.y * BM;
    if (m0 >= cnt) return;
    const int n0  = blockIdx.x * BN;
    const int tid = threadIdx.x;

    if (tid < BM) {
        tok[tid] = perm[e * CAP + m0 + tid];
        pr[tid]  = pprob[e * CAP + m0 + tid];
    }
    __syncthreads();

    // ---- per-thread copy lanes (fixed across K chunks): 4 b128 per chunk ----
    const int r0  = tid >> 2;             // rows 0..63
    const int r1  = (tid + 256) >> 2;     // rows 64..127
    const int cc  = (tid & 3) * 4;        // float offset within 16-float chunk
    const float* Wb  = We + ((size_t)e * DOUT + n0) * DIN;
    const float* gA0 = x  + (size_t)tok[r0] * DIN + cc;
    const float* gA1 = x  + (size_t)tok[r1] * DIN + cc;
    const float* gB0 = Wb + (size_t)r0 * DIN + cc;
    const float* gB1 = Wb + (size_t)r1 * DIN + cc;
    float* lA0 = As + r0 * LDA + cc;
    float* lA1 = As + r1 * LDA + cc;
    float* lB0 = Bs + r0 * LDA + cc;
    float* lB1 = Bs + r1 * LDA + cc;

    // ---- wave tiling ----
    const int lane = tid & 31;
    const int w    = tid >> 5;
    const int wm   = (w & 3) * 32;        // 4 wave-rows  -> 128 tokens
    const int wn   = (w >> 2) * 64;       // 2 wave-cols  -> 128 outs
    const int lr   = lane & 15;
    const int khi  = (lane >> 4) * 2;     // lanes 16..31 hold K+2,K+3

    v8f c[2][4];
#pragma unroll
    for (int mi = 0; mi < 2; ++mi)
#pragma unroll
        for (int ni = 0; ni < 4; ++ni)
            c[mi][ni] = (v8f){0.f, 0.f, 0.f, 0.f, 0.f, 0.f, 0.f, 0.f};

    const int nch = DIN / KC;             // 64 chunks
    // prefetch chunk 0 into buffer 0
    cp16(lA0, gA0); cp16(lA1, gA1); cp16(lB0, gB0); cp16(lB1, gB1);

    for (int ch = 0; ch < nch; ++ch) {
        const int cur = ch & 1;
        if (ch + 1 < nch) {               // issue next chunk, then wait for current
            const int nb = (cur ^ 1) * BUFE;
            const int k1 = (ch + 1) * KC;
            cp16(lA0 + nb, gA0 + k1); cp16(lA1 + nb, gA1 + k1);
            cp16(lB0 + nb, gB0 + k1); cp16(lB1 + nb, gB1 + k1);
            ASYNC_WAIT(4);                // in-order: current chunk has landed
        } else {
            ASYNC_WAIT(0);
        }
        __syncthreads();

        const float* Ap = As + cur * BUFE + (wm + lr) * LDA + khi;
        const float* Bp = Bs + cur * BUFE + (wn + lr) * LDA + khi;
#pragma unroll
        for (int kk = 0; kk < KC; kk += 4) {
            v2f a0 = *(const v2f*)(Ap + kk);
            v2f a1 = *(const v2f*)(Ap + 16 * LDA + kk);
            v2f b0 = *(const v2f*)(Bp + kk);
            v2f b1 = *(const v2f*)(Bp + 16 * LDA + kk);
            v2f b2 = *(const v2f*)(Bp + 32 * LDA + kk);
            v2f b3 = *(const v2f*)(Bp + 48 * LDA + kk);
            c[0][0] = __builtin_amdgcn_wmma_f32_16x16x4_f32(false, a0, false, b0, (short)0, c[0][0], false, false);
            c[0][1] = __builtin_amdgcn_wmma_f32_16x16x4_f32(false, a0, false, b1, (short)0, c[0][1], false, false);
            c[0][2] = __builtin_amdgcn_wmma_f32_16x16x4_f32(false, a0, false, b2, (short)0, c[0][2], false, false);
            c[0][3] = __builtin_amdgcn_wmma_f32_16x16x4_f32(false, a0, false, b3, (short)0, c[0][3], false, false);
            c[1][0] = __builtin_amdgcn_wmma_f32_16x16x4_f32(false, a1, false, b0, (short)0, c[1][0], false, false);
            c[1][1] = __builtin_amdgcn_wmma_f32_16x16x4_f32(false, a1, false, b1, (short)0, c[1][1], false, false);
            c[1][2] = __builtin_amdgcn_wmma_f32_16x16x4_f32(false, a1, false, b2, (short)0, c[1][2], false, false);
            c[1][3] = __builtin_amdgcn_wmma_f32_16x16x4_f32(false, a1, false, b3, (short)0, c[1][3], false, false);
        }
        __syncthreads();
    }

    // ---- epilogue: scale by gate prob, scatter-add into token rows ----
    const int colb  = n0 + wn + lr;
    const int mbase = wm + ((lane >> 4) << 3);   // VGPR i holds M=i (+8 for hi lanes)
#pragma unroll
    for (int mi = 0; mi < 2; ++mi) {
#pragma unroll
        for (int i = 0; i < 8; ++i) {
            const int m = mbase + mi * 16 + i;
            const float p = pr[m];
            float* orow = out + (size_t)tok[m] * DOUT;
#pragma unroll
            for (int ni = 0; ni < 4; ++ni)
                atomicAdd(orow + colb + ni * 16, c[mi][ni][i] * p);
        }
    }
}

// ---------------------------------------------------------------- aux loss
__global__ void k_aux(const float* __restrict__ eload, float* __restrict__ out_aux) {
    if (threadIdx.x == 0) {
        float s = 0.f, ss = 0.f;
        for (int e = 0; e < NEXP; ++e) { float v = eload[e]; s += v; ss += v * v; }
        float mean = s / (float)NEXP;
        float var  = (ss - (float)NEXP * mean * mean) / (float)(NEXP - 1);
        var = fmaxf(var, 0.0f);
        out_aux[0] = sqrtf(var) / mean;           // LAMBDA_AUX = 1
    }
}

// ---------------------------------------------------------------- host launcher
extern "C" void kernel_launch(void* const* d_in, const int* in_sizes, int n_in,
                              void* d_out, int out_size, void* d_ws, size_t ws_size,
                              hipStream_t stream) {
    const float* x  = (const float*)d_in[0];
    const float* Wg = (const float*)d_in[1];
    const float* bg = (const float*)d_in[2];
    const float* We = (const float*)d_in[3];
    const float* be = (const float*)d_in[4];
    float* out = (float*)d_out;

    char* ws = (char*)d_ws;
    int*   counts = (int*)  (ws + 0);
    float* eload  = (float*)(ws + 64);
    int*   tkidx  = (int*)  (ws + 128);
    float* tkp    = (float*)(ws + 128 + (size_t)BATCH * 2 * 4);
    int*   perm   = (int*)  (ws + 128 + (size_t)BATCH * 4 * 4);
    float* pprob  = (float*)(ws + 128 + (size_t)BATCH * 4 * 4 + (size_t)NEXP * CAP * 4);

    k_zero<<<1, 32, 0, stream>>>(counts, eload);
    k_gate<<<BATCH / 256, 256, 0, stream>>>(x, Wg, bg, tkidx, tkp,
                                            counts, eload, perm, pprob);
    k_pad<<<NEXP, 128, 0, stream>>>(counts, perm, pprob);
    k_init<<<BATCH, 256, 0, stream>>>(be, tkidx, tkp, out);

    dim3 grid(DOUT / BN, BATCH / BM, NEXP);
    k_gemm<<<grid, 256, 0, stream>>>(x, We, counts, perm, pprob, out);

    k_aux<<<1, 32, 0, stream>>>(eload, out + (size_t)BATCH * DOUT);
}